// DMRG_27676769255718
// MI455X (gfx1250) — compile-verified
//
#include <hip/hip_runtime.h>

typedef __attribute__((ext_vector_type(2))) float v2f;
typedef __attribute__((ext_vector_type(4))) float f4;
typedef __attribute__((ext_vector_type(8))) float v8f;

#define CHI 1024
#define MEG (CHI * CHI)

// ---------------------------------------------------------------------------
// CDNA5 async global->LDS copy (ASYNCcnt path), guarded so we always compile.
// Builtin parameter types (from hipcc diagnostic): gcc-style int4 pointers,
// global side in AS(1) ("__device__"), LDS side in AS(3).
// ---------------------------------------------------------------------------
#if __has_builtin(__builtin_amdgcn_global_load_async_to_lds_b128)
#define HAVE_ASYNC_LDS 1
typedef int v4i_b __attribute__((vector_size(16)));
__device__ __forceinline__ void cp_async16(const float* g, float* l) {
    __builtin_amdgcn_global_load_async_to_lds_b128(
        (__attribute__((address_space(1))) v4i_b*)g,
        (__attribute__((address_space(3))) v4i_b*)l, 0, 0);
}
#else
#define HAVE_ASYNC_LDS 0
#endif

#if __has_builtin(__builtin_amdgcn_s_wait_asynccnt)
#define WAIT_ASYNC() __builtin_amdgcn_s_wait_asynccnt(0)
#else
#define WAIT_ASYNC() asm volatile("s_wait_asynccnt 0" ::: "memory")
#endif

// ---------------------------------------------------------------------------
// Kernel 0: fuse the two MPO tensors.
// T[b][i][j][f][c][e] = sum_d M1[b,d,i,c] * M2[d,f,j,e]   (400 elements)
// ---------------------------------------------------------------------------
__global__ void dmrg_mpo_fuse(const float* __restrict__ M1,
                              const float* __restrict__ M2,
                              float* __restrict__ T) {
    int t = threadIdx.x + blockIdx.x * blockDim.x;
    if (t >= 400) return;
    int e = t & 1;
    int c = (t >> 1) & 1;
    int f = (t >> 2) % 5;
    int j = (t / 20) & 1;
    int i = (t / 40) & 1;
    int b = t / 80;
    float acc = 0.f;
#pragma unroll
    for (int d = 0; d < 5; ++d)
        acc += M1[((b * 5 + d) * 2 + i) * 2 + c] * M2[((d * 5 + f) * 2 + j) * 2 + e];
    T[t] = acc;
}

// ---------------------------------------------------------------------------
// Unified NT GEMM:  C[m][n] = sum_slices sum_k A[m][k] * B[n][k]
//   A pitch 1024 (K contiguous), B pitch 1024 (K contiguous), K per slice 1024.
//   Block tile 128x128, K-tile 32, 8 wave32s, wave tile 32(M)x64(N),
//   V_WMMA_F32_16X16X4_F32, double-buffered LDS, async B128 copies.
// Used for:
//   step 2:  W[f][(a,c,e)][k] = sum_g psi[(a,c,e)][g] * R[f][k][g]
//   step 4:  res[h][(i,j,k)]  = sum_b sum_a L[b][h][a] * Vt[b][(i,j,k)][a]
// ---------------------------------------------------------------------------
#define TPITCH 36   // LDS row pitch in floats (144 B rows -> 16B aligned cols)

__device__ __forceinline__ void tile_issue(const float* __restrict__ gbase,
                                           float* lds, int tid) {
#pragma unroll
    for (int q = 0; q < 4; ++q) {
        const int idx = q * 256 + tid;
        const int row = idx >> 3;          // 0..127
        const int col = (idx & 7) * 4;     // 0..28
#if HAVE_ASYNC_LDS
        cp_async16(gbase + (size_t)row * CHI + col, lds + row * TPITCH + col);
#else
        *(f4*)&lds[row * TPITCH + col] =
            *(const f4*)&gbase[(size_t)row * CHI + col];
#endif
    }
}

__global__ __launch_bounds__(256)
void dmrg_gemm_nt(const float* __restrict__ A, const float* __restrict__ B,
                  float* __restrict__ C, int ldc, int nSlices,
                  unsigned long long aSlice, unsigned long long bSlice,
                  unsigned long long aBatch, unsigned long long bBatch,
                  unsigned long long cBatch) {
    __shared__ float As[2][128 * TPITCH];
    __shared__ float Bs[2][128 * TPITCH];

    const float* __restrict__ Ab = A + (size_t)blockIdx.z * aBatch;
    const float* __restrict__ Bb = B + (size_t)blockIdx.z * bBatch;
    float* __restrict__ Cb = C + (size_t)blockIdx.z * cBatch;

    const int n0 = blockIdx.x * 128;
    const int m0 = blockIdx.y * 128;

    const int tid  = threadIdx.x;
    const int lane = tid & 31;
    const int wave = tid >> 5;
    const int wm   = wave & 3;    // 4 wave rows of 32 M
    const int wn   = wave >> 2;   // 2 wave cols of 64 N
    const int half = lane >> 4;   // lane half selects K pair / M+8 per ISA
    const int lr   = lane & 15;

    v8f acc[2][4];
#pragma unroll
    for (int mt = 0; mt < 2; ++mt)
#pragma unroll
        for (int nt = 0; nt < 4; ++nt)
            acc[mt][nt] = (v8f){0.f, 0.f, 0.f, 0.f, 0.f, 0.f, 0.f, 0.f};

    const int KT = nSlices * (CHI / 32);

#define ATILE(kt) (Ab + (size_t)((kt) >> 5) * aSlice + (size_t)m0 * CHI + ((kt) & 31) * 32)
#define BTILE(kt) (Bb + (size_t)((kt) >> 5) * bSlice + (size_t)n0 * CHI + ((kt) & 31) * 32)

    tile_issue(ATILE(0), As[0], tid);
    tile_issue(BTILE(0), Bs[0], tid);
#if HAVE_ASYNC_LDS
    WAIT_ASYNC();
#endif
    __syncthreads();

    for (int kt = 0; kt < KT; ++kt) {
        const int cur = kt & 1;
        if (kt + 1 < KT) {
            tile_issue(ATILE(kt + 1), As[cur ^ 1], tid);
            tile_issue(BTILE(kt + 1), Bs[cur ^ 1], tid);
        }
        const float* __restrict__ at = As[cur];
        const float* __restrict__ bt = Bs[cur];
#pragma unroll
        for (int ks = 0; ks < 8; ++ks) {
            const int kb = ks * 4 + 2 * half;   // this lane-half's K pair
            v2f a[2], bf[4];
#pragma unroll
            for (int mt = 0; mt < 2; ++mt) {
                int row = wm * 32 + mt * 16 + lr;
                a[mt] = *(const v2f*)&at[row * TPITCH + kb];
            }
#pragma unroll
            for (int nt = 0; nt < 4; ++nt) {
                int col = wn * 64 + nt * 16 + lr;
                bf[nt] = *(const v2f*)&bt[col * TPITCH + kb];
            }
#pragma unroll
            for (int mt = 0; mt < 2; ++mt)
#pragma unroll
                for (int nt = 0; nt < 4; ++nt)
                    acc[mt][nt] = __builtin_amdgcn_wmma_f32_16x16x4_f32(
                        false, a[mt], false, bf[nt], (short)0, acc[mt][nt],
                        false, false);
        }
#if HAVE_ASYNC_LDS
        WAIT_ASYNC();
#endif
        __syncthreads();
    }
#undef ATILE
#undef BTILE

    // D layout: VGPR r -> M = r + 8*half, N = lr
#pragma unroll
    for (int mt = 0; mt < 2; ++mt)
#pragma unroll
        for (int nt = 0; nt < 4; ++nt)
#pragma unroll
            for (int r = 0; r < 8; ++r) {
                int m = m0 + wm * 32 + mt * 16 + half * 8 + r;
                int n = n0 + wn * 64 + nt * 16 + lr;
                Cb[(size_t)m * ldc + n] = acc[mt][nt][r];
            }
}

// ---------------------------------------------------------------------------
// Kernel 2: Vt[b][(i,j,k)][a] = sum_{f,c,e} T[b,i,j,f,c,e] * W[f][(a,c,e)][k]
// 32x32 (a,k) block with an LDS transpose so both the W reads (k-contiguous)
// and the Vt writes (a-contiguous) are coalesced. 0.84 GFLOP total.
// ---------------------------------------------------------------------------
__global__ __launch_bounds__(256)
void dmrg_contract_T(const float* __restrict__ Tg,
                     const float* __restrict__ Wbuf,
                     float* __restrict__ Vt) {
    __shared__ float sT[400];
    __shared__ float Lt[20][32][33];

    const int tid = threadIdx.x;
    for (int t = tid; t < 400; t += 256) sT[t] = Tg[t];
    __syncthreads();

    const int k0 = blockIdx.x * 32;
    const int a0 = blockIdx.y * 32;

    // phase 1: compute, k-contiguous reads, store transposable tile in LDS
    {
        const int kl = tid & 31;
        const int ab = tid >> 5;   // 0..7
#pragma unroll
        for (int ai = 0; ai < 4; ++ai) {
            const int al = ab + ai * 8;
            const int a  = a0 + al;
            float w[5][2][2];
#pragma unroll
            for (int f = 0; f < 5; ++f)
#pragma unroll
                for (int c = 0; c < 2; ++c)
#pragma unroll
                    for (int e = 0; e < 2; ++e)
                        w[f][c][e] = Wbuf[(size_t)f * (4096ull * CHI) +
                                          (size_t)(a * 4 + c * 2 + e) * CHI +
                                          k0 + kl];
#pragma unroll
            for (int o = 0; o < 20; ++o) {   // o = b*4 + i*2 + j
                float acc = 0.f;
#pragma unroll
                for (int f = 0; f < 5; ++f)
#pragma unroll
                    for (int c = 0; c < 2; ++c)
#pragma unroll
                        for (int e = 0; e < 2; ++e)
                            acc += sT[(o * 5 + f) * 4 + c * 2 + e] * w[f][c][e];
                Lt[o][kl][al] = acc;
            }
        }
    }
    __syncthreads();

    // phase 2: a-contiguous writes
    {
        const int al = tid & 31;
        const int kb = tid >> 5;
#pragma unroll
        for (int ki = 0; ki < 4; ++ki) {
            const int kl = kb + ki * 8;
#pragma unroll
            for (int o = 0; o < 20; ++o) {
                const int b  = o >> 2;
                const int ij = o & 3;
                Vt[(size_t)b * (4096ull * CHI) +
                   (size_t)(ij * CHI + k0 + kl) * CHI + a0 + al] = Lt[o][kl][al];
            }
        }
    }
}

// ---------------------------------------------------------------------------
extern "C" void kernel_launch(void* const* d_in, const int* in_sizes, int n_in,
                              void* d_out, int out_size, void* d_ws, size_t ws_size,
                              hipStream_t stream) {
    const float* psi = (const float*)d_in[0];
    const float* L   = (const float*)d_in[1];
    const float* M1  = (const float*)d_in[2];
    const float* M2  = (const float*)d_in[3];
    const float* R   = (const float*)d_in[4];
    float* out = (float*)d_out;

    float* ws   = (float*)d_ws;
    float* Tg   = ws;                        // 400 floats (pad to 1024)
    float* Wbuf = ws + 1024;                 // 5*4096*1024 floats
    float* Vt   = Wbuf + 5ull * 4096 * CHI;  // 5*4096*1024 floats
    // total workspace ~167.8 MB

    dmrg_mpo_fuse<<<2, 256, 0, stream>>>(M1, M2, Tg);

    // step 2: W[f] = psi (4096x1024) x R[f]^T (1024x1024)
    dmrg_gemm_nt<<<dim3(8, 32, 5), 256, 0, stream>>>(
        psi, R, Wbuf, /*ldc*/ CHI, /*nSlices*/ 1,
        /*aSlice*/ 0ull, /*bSlice*/ 0ull,
        /*aBatch*/ 0ull, /*bBatch*/ (unsigned long long)MEG,
        /*cBatch*/ 4096ull * CHI);

    // step 3: Vt[b][(i,j,k)][a]
    dmrg_contract_T<<<dim3(32, 32), 256, 0, stream>>>(Tg, Wbuf, Vt);

    // step 4: out (1024x4096) = sum_b L[b] (1024x1024) x Vt[b]^T (1024x4096)
    dmrg_gemm_nt<<<dim3(32, 8, 1), 256, 0, stream>>>(
        L, Vt, out, /*ldc*/ 4096, /*nSlices*/ 5,
        /*aSlice*/ (unsigned long long)MEG, /*bSlice*/ 4096ull * CHI,
        /*aBatch*/ 0ull, /*bBatch*/ 0ull, /*cBatch*/ 0ull);
}